// PointConvMessage_34291018891264
// MI455X (gfx1250) — compile-verified
//
#include <hip/hip_runtime.h>
#include <hip/hip_bf16.h>

typedef float v2f __attribute__((ext_vector_type(2)));
typedef float v8f __attribute__((ext_vector_type(8)));

#define C_S3   0.57735026918962576f   // 3^-0.5
#define C_S2   0.70710678118654752f   // 2^-0.5
#define C_S2S3 0.40824829046386302f   // 2^-0.5 * 3^-0.5
#define C_SMUL 0.17677669529663689f   // 32^-0.5
#define C_SEMB 0.25f                  // 16^-0.5
#define C_SHID 0.35355339059327376f   // 8^-0.5

__device__ __forceinline__ float silu_f(float x) {
    return x / (1.0f + __expf(-x));
}

// ---------------------------------------------------------------------------
// Kernel 1: per-node linear transforms.
//   y[n][c]            c<32 : y0 = (x0 @ W0) * 32^-0.5
//   y[n][32+v*3+i]     else : y1[v][i] = sum_u x1[u][i]*W1[u][v] * 32^-0.5
// One wave per node; x row staged in wave-private LDS; 4 cols per lane.
// ---------------------------------------------------------------------------
__global__ void node_linear_kernel(const float* __restrict__ nf,
                                   const float* __restrict__ w0,
                                   const float* __restrict__ w1,
                                   float* __restrict__ y, int nN)
{
    __shared__ float xsh[4][128];
    const int lane = threadIdx.x & 31;
    const int wv   = threadIdx.x >> 5;
    const int n    = blockIdx.x * 4 + wv;
    if (n >= nN) return;

    float* xs = xsh[wv];
    float4 xv = reinterpret_cast<const float4*>(nf + (size_t)n * 128)[lane];
    xs[lane * 4 + 0] = xv.x; xs[lane * 4 + 1] = xv.y;
    xs[lane * 4 + 2] = xv.z; xs[lane * 4 + 3] = xv.w;
    asm volatile("" ::: "memory");   // same-wave DS ops are in-order in HW

    float4 o;
    float* op = &o.x;
#pragma unroll
    for (int q = 0; q < 4; ++q) {
        const int c = lane * 4 + q;
        float acc = 0.0f;
        if (c < 32) {
#pragma unroll
            for (int u = 0; u < 32; ++u) acc += xs[u] * w0[u * 32 + c];
        } else {
            const int r = c - 32;
            const int m = r / 3, i = r - 3 * m;
#pragma unroll
            for (int u = 0; u < 32; ++u) acc += xs[32 + u * 3 + i] * w1[u * 32 + m];
        }
        op[q] = acc * C_SMUL;
    }
    reinterpret_cast<float4*>(y + (size_t)n * 128)[lane] = o;
}

// ---------------------------------------------------------------------------
// Kernel 2: edge MLP via V_WMMA_F32_16X16X4_F32 + gather + message assembly.
// 128 threads = 4 waves, 16 edges per wave (64 edges / block).
// ---------------------------------------------------------------------------
__global__ void edge_message_kernel(const float* __restrict__ attrs,
                                    const float* __restrict__ emb,
                                    const int*   __restrict__ esrc,
                                    const float* __restrict__ w1p,  // mlp_w1: 16x8
                                    const float* __restrict__ w2p,  // mlp_w2: 8x128
                                    const float* __restrict__ y,
                                    float* __restrict__ msg, int nE)
{
    __shared__ float lds[4 * 16 * 128];          // 32 KB: per-wave 16x128 w block
    const int lane = threadIdx.x & 31;
    const int wv   = threadIdx.x >> 5;
    const int mrow = lane & 15;                  // M (edge) for A, N for B/C
    const int koff = (lane >= 16) ? 2 : 0;       // K sub-offset for hi half-wave
    const int hiM  = (lane >= 16) ? 8 : 0;       // C-matrix M offset for hi half
    float* hl = lds + wv * 2048;

    const int eb = blockIdx.x * 64 + wv * 16;

    // speculative prefetch of the next tile's embeddings -> global_prefetch_b8
    __builtin_prefetch(emb + (size_t)(eb + 64) * 16, 0, 1);

    // ---- MLP layer 1: h = silu( (emb 16x16) @ (w1 padded 16x16) * 16^-0.5 )
    int eA = eb + mrow; if (eA >= nE) eA = nE - 1;   // clamp, keep EXEC full
    const float* ar = emb + (size_t)eA * 16;

    // zero-pad B cols 8..15 with a multiply mask (no divergent loads)
    const float bmask = (mrow < 8) ? 1.0f : 0.0f;
    const int   bn    = mrow & 7;                    // always in-bounds

    v8f c = {};
#pragma unroll
    for (int s = 0; s < 4; ++s) {
        const int k = 4 * s + koff;
        v2f a = { ar[k], ar[k + 1] };
        v2f b = { w1p[k * 8 + bn] * bmask,
                  w1p[(k + 1) * 8 + bn] * bmask };
        c = __builtin_amdgcn_wmma_f32_16x16x4_f32(false, a, false, b,
                                                  (short)0, c, false, false);
    }
#pragma unroll
    for (int j = 0; j < 8; ++j) c[j] = silu_f(c[j] * C_SEMB);

    // h (16x16, cols>=8 are zero) -> wave LDS in C layout, re-read in A layout
#pragma unroll
    for (int j = 0; j < 8; ++j) hl[(j + hiM) * 16 + mrow] = c[j];
    asm volatile("" ::: "memory");
    v2f a2[2];
#pragma unroll
    for (int s = 0; s < 2; ++s) {
        const int k = 4 * s + koff;
        a2[s] = { hl[mrow * 16 + k], hl[mrow * 16 + k + 1] };
    }
    asm volatile("" ::: "memory");

    // ---- MLP layer 2: w(16x128) = (h 16x8) @ (w2 8x128) * 8^-0.5  -> LDS
#pragma unroll
    for (int t = 0; t < 8; ++t) {
        v8f cw = {};
#pragma unroll
        for (int s = 0; s < 2; ++s) {
            const int k = 4 * s + koff;
            v2f b = { w2p[k * 128 + 16 * t + mrow],
                      w2p[(k + 1) * 128 + 16 * t + mrow] };
            cw = __builtin_amdgcn_wmma_f32_16x16x4_f32(false, a2[s], false, b,
                                                       (short)0, cw, false, false);
        }
#pragma unroll
        for (int j = 0; j < 8; ++j)
            hl[(j + hiM) * 128 + 16 * t + mrow] = cw[j] * C_SHID;
    }
    asm volatile("" ::: "memory");

    // ---- per-edge message assembly: 4 output cols per lane, float4 stores
    for (int el = 0; el < 16; ++el) {
        const int e  = eb + el;
        const int ec = (e < nE) ? e : (nE - 1);
        const int src = esrc[ec];
        const float* yr = y + (size_t)src * 128;
        const float4 at = reinterpret_cast<const float4*>(attrs)[ec];
        const float a0 = at.x, a10 = at.y, a11 = at.z, a12 = at.w;
        const float* wrow = hl + el * 128;

        // diagonal y values yr[4*lane .. 4*lane+3] == yr[cc] for q=0..3
        float4 yc = reinterpret_cast<const float4*>(yr)[lane];
        const float* ycp = &yc.x;

        float4 o;
        float* op = &o.x;
#pragma unroll
        for (int q = 0; q < 4; ++q) {
            const int cc = lane * 4 + q;
            const float ycq = ycp[q];            // == yr[cc]
            float out;
            if (cc < 32) {
                const int m = cc;                // yr[m] == ycq here
                const float d = yr[32 + 3 * m] * a10 + yr[32 + 3 * m + 1] * a11
                              + yr[32 + 3 * m + 2] * a12;
                out = C_S2 * (wrow[m] * ycq * a0 + C_S3 * wrow[32 + m] * d);
            } else {
                const int r = cc - 32;
                const int m = r / 3, i = r - 3 * m;
                const float ai = (i == 0) ? a10 : ((i == 1) ? a11 : a12);
                out = C_S2S3 * (wrow[64 + m] * yr[m] * ai
                              + wrow[96 + m] * ycq * a0);
            }
            op[q] = out;
        }
        if (e < nE)
            reinterpret_cast<float4*>(msg + (size_t)e * 128)[lane] = o;
    }
}

extern "C" void kernel_launch(void* const* d_in, const int* in_sizes, int n_in,
                              void* d_out, int out_size, void* d_ws, size_t ws_size,
                              hipStream_t stream) {
    const float* nf    = (const float*)d_in[0];   // node_feats  (N, 128)
    const float* attrs = (const float*)d_in[1];   // edge_attrs  (E, 4)
    const float* emb   = (const float*)d_in[2];   // edge_embed  (E, 16)
    const int*   esrc  = (const int*)  d_in[3];   // edge_src    (E,)
    /* d_in[4] = edge_dst : unused by the reference */
    const float* w0    = (const float*)d_in[5];   // lin_w0 (32,32)
    const float* w1    = (const float*)d_in[6];   // lin_w1 (32,32)
    const float* mw1   = (const float*)d_in[7];   // mlp_w1 (16,8)
    const float* mw2   = (const float*)d_in[8];   // mlp_w2 (8,128)
    float* msg = (float*)d_out;
    float* yws = (float*)d_ws;                    // y table: nN*128 floats (~25.6 MB)

    const int nN = in_sizes[0] / 128;
    const int nE = in_sizes[3];

    const int b1 = (nN + 3) / 4;                  // one wave per node
    node_linear_kernel<<<b1, 128, 0, stream>>>(nf, w0, w1, yws, nN);

    const int b2 = (nE + 63) / 64;                // 64 edges per block
    edge_message_kernel<<<b2, 128, 0, stream>>>(attrs, emb, esrc, mw1, mw2,
                                                yws, msg, nE);
}